// Gemma3nAltUp_4990751998313
// MI455X (gfx1250) — compile-verified
//
#include <hip/hip_runtime.h>

#define HDIM 2048
#define ADIM 4
#define EPSV 1e-6f
#define CLIPV 120.0f

typedef __attribute__((ext_vector_type(2))) float v2f;
typedef __attribute__((ext_vector_type(4))) float f4;
typedef __attribute__((ext_vector_type(8))) float v8f;

__device__ inline float clipf(float v) { return fminf(fmaxf(v, -CLIPV), CLIPV); }

// ---------------------------------------------------------------------------
// Kernel 1: per-token router math.
// Block = 512 threads = 16 waves; wave w handles token (blockIdx.x*16 + w):
//   coalesced reduction over H of sum(x^2) and 4 dots x . (rnw * w_router[a]).
// Then wave 0 computes, for all 16 tokens of the block at once:
//   coef[16x16]  = m_pred @ clip(w_pred).T   via V_WMMA_F32_16X16X4_F32
//   cc  [16x4]   = m_corr @ clip(w_corr).T+1 via V_WMMA_F32_16X16X4_F32 (B padded)
// ---------------------------------------------------------------------------
__global__ void __launch_bounds__(512) altup_router_kernel(
    const float* __restrict__ hs0,   // hidden_states[0] : [T, H]
    const float* __restrict__ act,   // activated        : [T, H]
    const float* __restrict__ rnw,   // router_norm_weight [H]
    const float* __restrict__ wr,    // w_router [4, H]
    const float* __restrict__ wp,    // w_pred   [16, 4]
    const float* __restrict__ wc,    // w_corr   [4, 4]
    float* __restrict__ coef_ws,     // [T, 16]  coef[t][b*4+a] = all_coefs[t,a,b]
    float* __restrict__ cc_ws)       // [T, 4]
{
    __shared__ float s_mp[16][4];
    __shared__ float s_mc[16][4];

    const int wave = threadIdx.x >> 5;   // 0..15
    const int lane = threadIdx.x & 31;
    const int tok_base = blockIdx.x * 16;
    const int t = tok_base + wave;

    const float* xp = hs0 + (size_t)t * HDIM;
    const float* xc = act + (size_t)t * HDIM;

    float ssp = 0.f, ssc = 0.f;
    float dp0 = 0.f, dp1 = 0.f, dp2 = 0.f, dp3 = 0.f;
    float dc0 = 0.f, dc1 = 0.f, dc2 = 0.f, dc3 = 0.f;

    for (int h = lane; h < HDIM; h += 32) {
        float a = xp[h];
        float b = xc[h];
        float g  = rnw[h];
        float g0 = g * wr[h];
        float g1 = g * wr[HDIM + h];
        float g2 = g * wr[2 * HDIM + h];
        float g3 = g * wr[3 * HDIM + h];
        ssp = fmaf(a, a, ssp);  ssc = fmaf(b, b, ssc);
        dp0 = fmaf(a, g0, dp0); dp1 = fmaf(a, g1, dp1);
        dp2 = fmaf(a, g2, dp2); dp3 = fmaf(a, g3, dp3);
        dc0 = fmaf(b, g0, dc0); dc1 = fmaf(b, g1, dc1);
        dc2 = fmaf(b, g2, dc2); dc3 = fmaf(b, g3, dc3);
    }
    // wave32 butterfly reduction
    #pragma unroll
    for (int off = 16; off >= 1; off >>= 1) {
        ssp += __shfl_xor(ssp, off, 32); ssc += __shfl_xor(ssc, off, 32);
        dp0 += __shfl_xor(dp0, off, 32); dp1 += __shfl_xor(dp1, off, 32);
        dp2 += __shfl_xor(dp2, off, 32); dp3 += __shfl_xor(dp3, off, 32);
        dc0 += __shfl_xor(dc0, off, 32); dc1 += __shfl_xor(dc1, off, 32);
        dc2 += __shfl_xor(dc2, off, 32); dc3 += __shfl_xor(dc3, off, 32);
    }
    if (lane == 0) {
        const float inv_h = 1.0f / (float)HDIM;
        float sp = __frsqrt_rn(ssp * inv_h + EPSV) * inv_h;
        float sc = __frsqrt_rn(ssc * inv_h + EPSV) * inv_h;
        s_mp[wave][0] = tanhf(dp0 * sp); s_mp[wave][1] = tanhf(dp1 * sp);
        s_mp[wave][2] = tanhf(dp2 * sp); s_mp[wave][3] = tanhf(dp3 * sp);
        s_mc[wave][0] = tanhf(dc0 * sc); s_mc[wave][1] = tanhf(dc1 * sc);
        s_mc[wave][2] = tanhf(dc2 * sc); s_mc[wave][3] = tanhf(dc3 * sc);
    }
    __syncthreads();

    if (wave == 0) {
        // A operand (16x4 f32): lanes 0-15 rows, VGPR0/1 = K pair (K0,K1);
        // lanes 16-31 same rows, K pair (K2,K3).
        const int m = lane & 15;
        const int kh = lane >> 4;       // 0 -> K{0,1}, 1 -> K{2,3}
        v2f ap, ac, bp, bc;
        ap.x = s_mp[m][kh * 2 + 0]; ap.y = s_mp[m][kh * 2 + 1];
        ac.x = s_mc[m][kh * 2 + 0]; ac.y = s_mc[m][kh * 2 + 1];
        // B operand (4x16 f32): B(k, n) = clip(w)[n][k]; n = lane&15, K per kh.
        const int n = m;
        bp.x = clipf(wp[n * 4 + kh * 2 + 0]);
        bp.y = clipf(wp[n * 4 + kh * 2 + 1]);
        bc.x = (n < 4) ? clipf(wc[n * 4 + kh * 2 + 0]) : 0.0f;
        bc.y = (n < 4) ? clipf(wc[n * 4 + kh * 2 + 1]) : 0.0f;

        v8f z = {};
        v8f dcoef = __builtin_amdgcn_wmma_f32_16x16x4_f32(
            false, ap, false, bp, (short)0, z, false, false);
        v8f dcc = __builtin_amdgcn_wmma_f32_16x16x4_f32(
            false, ac, false, bc, (short)0, z, false, false);

        // D layout: VGPR r -> row r (lanes 0-15) / row r+8 (lanes 16-31), col n.
        #pragma unroll
        for (int r = 0; r < 8; ++r) {
            const int tok = tok_base + r + kh * 8;
            coef_ws[(size_t)tok * 16 + n] = dcoef[r];
            if (n < 4) cc_ws[(size_t)tok * 4 + n] = dcc[r] + 1.0f;
        }
    }
}

// ---------------------------------------------------------------------------
// Kernel 2: fused streaming mix. One block per token, 512 threads x float4.
// Per-token coefs are block-uniform -> scalar loads. All big traffic is
// coalesced 16B vector loads/stores.
//   pred[b]      = hs[b] + sum_a coef[b*4+a] * hs[a]
//   innov        = act - pred[0]
//   corrected[b] = innov * cc[b] + pred[b]
//   output       = corrected[0] * scale
// d_out = corrected [4,T,H] then output [T,H].
// ---------------------------------------------------------------------------
__global__ void __launch_bounds__(512) altup_mix_kernel(
    const float* __restrict__ hs,     // [4, T, H]
    const float* __restrict__ act,    // [T, H]
    const float* __restrict__ scale,  // [H]
    const float* __restrict__ coef,   // [T, 16]
    const float* __restrict__ cc,     // [T, 4]
    float* __restrict__ out,          // [5 * T * H]
    int T)
{
    const int t = blockIdx.x;
    const size_t TH = (size_t)T * HDIM;
    const size_t base = (size_t)t * HDIM + (size_t)threadIdx.x * 4;

    float c[16];
    #pragma unroll
    for (int i = 0; i < 16; ++i) c[i] = coef[(size_t)t * 16 + i];
    float ccv[4];
    #pragma unroll
    for (int b = 0; b < 4; ++b) ccv[b] = cc[(size_t)t * 4 + b];

    f4 x[4];
    #pragma unroll
    for (int a = 0; a < 4; ++a) x[a] = *(const f4*)(hs + (size_t)a * TH + base);
    f4 av = *(const f4*)(act + base);
    f4 sc = *(const f4*)(scale + (size_t)threadIdx.x * 4);

    f4 p[4];
    #pragma unroll
    for (int b = 0; b < 4; ++b) {
        f4 acc = x[b];
        #pragma unroll
        for (int a = 0; a < 4; ++a) acc += c[b * 4 + a] * x[a];
        p[b] = acc;
    }
    f4 innov = av - p[0];
    f4 corr0 = innov * ccv[0] + p[0];
    *(f4*)(out + base) = corr0;
    #pragma unroll
    for (int b = 1; b < 4; ++b) {
        f4 cr = innov * ccv[b] + p[b];
        *(f4*)(out + (size_t)b * TH + base) = cr;
    }
    *(f4*)(out + (size_t)4 * TH + base) = corr0 * sc;
}

extern "C" void kernel_launch(void* const* d_in, const int* in_sizes, int n_in,
                              void* d_out, int out_size, void* d_ws, size_t ws_size,
                              hipStream_t stream) {
    const float* hs    = (const float*)d_in[0];  // [4, T, H]
    const float* act   = (const float*)d_in[1];  // [T, H]
    const float* rnw   = (const float*)d_in[2];  // [H]
    const float* wr    = (const float*)d_in[3];  // [4, H]
    const float* wp    = (const float*)d_in[4];  // [16, 4]
    const float* wc    = (const float*)d_in[5];  // [4, 4]
    const float* cos_s = (const float*)d_in[6];  // [H]

    const int T = in_sizes[1] / HDIM;            // activated is [T, H]

    float* coef_ws = (float*)d_ws;               // [T, 16]
    float* cc_ws   = coef_ws + (size_t)T * 16;   // [T, 4]

    altup_router_kernel<<<T / 16, 512, 0, stream>>>(
        hs /* hidden_states[0] */, act, rnw, wr, wp, wc, coef_ws, cc_ws);

    altup_mix_kernel<<<T, 512, 0, stream>>>(
        hs, act, cos_s, coef_ws, cc_ws, (float*)d_out, T);
}